// ContrastiveLossOptimized_78451872629247
// MI455X (gfx1250) — compile-verified
//
#include <hip/hip_runtime.h>

typedef float v2f __attribute__((ext_vector_type(2)));
typedef float v8f __attribute__((ext_vector_type(8)));

#define TPB 256
#define BLOCKS 8
// waves = BLOCKS * TPB / 32 = 64; each wave consumes 64 elements per iteration.

__global__ void cl_init_ws(double* acc) {
    if (threadIdx.x < 4) acc[threadIdx.x] = 0.0;
}

__global__ __launch_bounds__(TPB)
void cl_reduce(const float* __restrict__ y_pred,
               const float* __restrict__ y_true,
               const int*   __restrict__ epoch,
               double* __restrict__ acc, int n) {
    const int lane          = threadIdx.x & 31;
    const int wavesPerBlock = TPB >> 5;
    const int waveId        = blockIdx.x * wavesPerBlock + (threadIdx.x >> 5);
    const int numWaves      = gridDim.x * wavesPerBlock;

    const float2* __restrict__ yp2 = (const float2*)y_pred;
    const float2* __restrict__ yt2 = (const float2*)y_true;
    const int nHalf = n >> 1;   // n is even (N = 16384)

    v8f accP  = {};   // sum p        (epoch >= 5)
    v8f accP2 = {};   // sum p^2      (epoch >= 5)
    v8f accT  = {};   // sum t        (epoch >= 5)
    v8f accB  = {};   // sum bce term (epoch < 5)
    const v2f ones = {1.0f, 1.0f};

    const bool need_bce = (*epoch < 5);   // scalar load -> uniform branch, EXEC stays all-1s

    if (!need_bce) {
        // ---- hot path (epoch >= 5): needs only S1, S2, T ----
        for (int base = waveId * 64; base < n; base += numWaves * 64) {
            const int ei   = base + 2 * lane;
            int       vidx = ei >> 1;
            if (vidx > nHalf - 1) vidx = nHalf - 1;       // clamp: load always legal
            const float m0 = (ei     < n) ? 1.0f : 0.0f;  // branchless validity masks
            const float m1 = (ei + 1 < n) ? 1.0f : 0.0f;

            const float2 xv = yp2[vidx];
            const float2 tv = yt2[vidx];

            // sigmoid via v_exp + v_rcp (masked so padded lanes contribute 0)
            const float p0 = m0 * __builtin_amdgcn_rcpf(1.0f + __expf(-xv.x));
            const float p1 = m1 * __builtin_amdgcn_rcpf(1.0f + __expf(-xv.y));

            const v2f ap  = {p0, p1};
            const v2f ap2 = {p0 * p0, p1 * p1};
            const v2f at  = {tv.x * m0, tv.y * m1};

            // D = A * ones(4x16) + C : every A element lands in a row-sum once.
            accP  = __builtin_amdgcn_wmma_f32_16x16x4_f32(false, ap,  false, ones, (short)0, accP,  false, false);
            accP2 = __builtin_amdgcn_wmma_f32_16x16x4_f32(false, ap2, false, ones, (short)0, accP2, false, false);
            accT  = __builtin_amdgcn_wmma_f32_16x16x4_f32(false, at,  false, ones, (short)0, accT,  false, false);
        }
    } else {
        // ---- epoch < 5: needs only the BCE sum ----
        for (int base = waveId * 64; base < n; base += numWaves * 64) {
            const int ei   = base + 2 * lane;
            int       vidx = ei >> 1;
            if (vidx > nHalf - 1) vidx = nHalf - 1;
            const float m0 = (ei     < n) ? 1.0f : 0.0f;
            const float m1 = (ei + 1 < n) ? 1.0f : 0.0f;

            const float2 xv = yp2[vidx];
            const float2 tv = yt2[vidx];
            const float x0 = xv.x, x1 = xv.y;

            // max(x,0) - x*t + log1p(exp(-|x|)); log1p(e) ~ __logf(1+e), e in (0,1]
            const float b0 = m0 * (fmaxf(x0, 0.0f) - x0 * tv.x
                                   + __logf(1.0f + __expf(-fabsf(x0))));
            const float b1 = m1 * (fmaxf(x1, 0.0f) - x1 * tv.y
                                   + __logf(1.0f + __expf(-fabsf(x1))));

            const v2f ab = {b0, b1};
            accB = __builtin_amdgcn_wmma_f32_16x16x4_f32(false, ab, false, ones, (short)0, accB, false, false);
        }
    }

    // Per-lane fold of the 8 accumulator rows, then full wave32 fold.
    float sP = 0.0f, sP2 = 0.0f, sT = 0.0f, sB = 0.0f;
#pragma unroll
    for (int v = 0; v < 8; ++v) {
        sP += accP[v]; sP2 += accP2[v]; sT += accT[v]; sB += accB[v];
    }
#pragma unroll
    for (int off = 16; off > 0; off >>= 1) {
        sP  += __shfl_xor(sP,  off, 32);
        sP2 += __shfl_xor(sP2, off, 32);
        sT  += __shfl_xor(sT,  off, 32);
        sB  += __shfl_xor(sB,  off, 32);
    }
    // Every column of D carries the same partial sum -> whole-matrix sum = 16*S.
    if (lane == 0) {
        atomicAdd(&acc[0], (double)(sP  * 0.0625f));
        atomicAdd(&acc[1], (double)(sP2 * 0.0625f));
        atomicAdd(&acc[2], (double)(sT  * 0.0625f));
        atomicAdd(&acc[3], (double)(sB  * 0.0625f));
    }
}

__global__ void cl_finalize(const double* __restrict__ acc,
                            const int* __restrict__ epoch,
                            float* __restrict__ out, int n) {
    if (threadIdx.x == 0 && blockIdx.x == 0) {
        const double S1 = acc[0], S2 = acc[1], T = acc[2], B = acc[3];
        const double dn = (double)n;
        double res;
        if (*epoch < 5) {
            res = B / dn;                      // mean BCE-with-logits
        } else {
            // sum_{i,j} loss = 2N*S2 - 2*S1^2 + 2N*T - 2*T^2 (cross terms cancel)
            res = 2.0 * dn * S2 - 2.0 * S1 * S1 + 2.0 * dn * T - 2.0 * T * T;
        }
        out[0] = (float)res;
    }
}

extern "C" void kernel_launch(void* const* d_in, const int* in_sizes, int n_in,
                              void* d_out, int out_size, void* d_ws, size_t ws_size,
                              hipStream_t stream) {
    const float* y_pred = (const float*)d_in[0];
    const float* y_true = (const float*)d_in[1];
    const int*   epoch  = (const int*)d_in[2];
    float* out = (float*)d_out;
    double* acc = (double*)d_ws;
    const int n = in_sizes[0];

    cl_init_ws<<<1, 32, 0, stream>>>(acc);
    cl_reduce<<<BLOCKS, TPB, 0, stream>>>(y_pred, y_true, epoch, acc, n);
    cl_finalize<<<1, 1, 0, stream>>>(acc, epoch, out, n);
}